// GCN_5841155522621
// MI455X (gfx1250) — compile-verified
//
#include <hip/hip_runtime.h>
#include <math.h>

typedef float v2f __attribute__((ext_vector_type(2)));
typedef float v8f __attribute__((ext_vector_type(8)));

#define DIM 128
#define RC_F 5.0f
#define XS_STRIDE 130   // pad 16x128 tile rows to spread LDS banks

// ---------------------------------------------------------------------------
// f[e] = exp(-(r-rs)^2/sigma^2) * 0.5*cos(pi*r/rc) * (r < rc)
// ---------------------------------------------------------------------------
__global__ void envelope_kernel(const float* __restrict__ e,
                                const float* __restrict__ rs,
                                const float* __restrict__ sigma,
                                float* __restrict__ f, int E) {
  int i = blockIdx.x * blockDim.x + threadIdx.x;
  if (i >= E) return;
  float r = e[i];
  float d = r - rs[0];
  float s = sigma[0];
  float g = __expf(-(d * d) / (s * s));
  float cut = 0.5f * __cosf(3.14159265358979323846f * r / RC_F);
  cut = (r < RC_F) ? cut : 0.0f;
  f[i] = g * cut;
}

__global__ void zero_kernel(float* __restrict__ p, int n) {
  int i = blockIdx.x * blockDim.x + threadIdx.x;
  if (i < n) p[i] = 0.0f;
}

// ---------------------------------------------------------------------------
// One wave (32 lanes) per edge; lane l handles features [4l, 4l+4).
// Coalesced 512B gather of cur[src], scaled by f[e], atomic-add into svf[dst].
// ---------------------------------------------------------------------------
__global__ void scatter_kernel(const float* __restrict__ cur,
                               const float* __restrict__ f,
                               const int* __restrict__ src,
                               const int* __restrict__ dst,
                               float* __restrict__ svf, int E) {
  int gid = blockIdx.x * blockDim.x + threadIdx.x;
  int edge = gid >> 5;
  int lane = gid & 31;
  if (edge >= E) return;
  int s = src[edge];
  int d = dst[edge];
  float w = f[edge];
  float4 m = ((const float4*)(cur + (size_t)s * DIM))[lane];
  float* op = svf + (size_t)d * DIM + lane * 4;
  unsafeAtomicAdd(op + 0, w * m.x);
  unsafeAtomicAdd(op + 1, w * m.y);
  unsafeAtomicAdd(op + 2, w * m.z);
  unsafeAtomicAdd(op + 3, w * m.w);
}

// ---------------------------------------------------------------------------
// out = relu((svf + v) @ W^T + b), fp32 WMMA 16x16x4.
// Block = 256 thr = 8 waves; block covers 16 nodes, wave w covers output
// features [16w, 16w+16). X tile staged in LDS once, shared by all 8 waves.
//
// A-matrix (16x4 f32): lane l, elem e -> M = l%16, K = kb + 2*(l/16) + e
// B-matrix (4x16 f32): lane l, elem e -> N = l%16, K = kb + 2*(l/16) + e
//   (B[k][n] = W[n_global][k], i.e. W^T slice)
// D (16x16 f32): lane l, vgpr r -> M = r + 8*(l/16), N = l%16
// ---------------------------------------------------------------------------
__global__ __launch_bounds__(256) void gemm_relu_kernel(
    const float* __restrict__ svf, const float* __restrict__ v,
    const float* __restrict__ W, const float* __restrict__ bias,
    float* __restrict__ out, int N) {
  __shared__ float Xs[16 * XS_STRIDE];
  const int tid = threadIdx.x;
  const int nb = blockIdx.x * 16;

  // cooperative load of X = svf + v (2048 floats, float4 granularity)
  for (int i = tid; i < (16 * DIM) / 4; i += 256) {
    int r = (i * 4) / DIM;
    int c = (i * 4) % DIM;
    int row = nb + r;
    row = row < N ? row : N - 1;
    const float4 sv = *(const float4*)(svf + (size_t)row * DIM + c);
    const float4 vv = *(const float4*)(v + (size_t)row * DIM + c);
    float* xp = Xs + r * XS_STRIDE + c;
    xp[0] = sv.x + vv.x;
    xp[1] = sv.y + vv.y;
    xp[2] = sv.z + vv.z;
    xp[3] = sv.w + vv.w;
  }
  __syncthreads();

  const int wave = tid >> 5;
  const int lane = tid & 31;
  const int hl = lane >> 4;   // 0 or 1
  const int m = lane & 15;
  const int jcol = wave * 16 + m;          // output feature / W row
  const float* wp = W + (size_t)jcol * DIM;

  v8f acc = {0.f, 0.f, 0.f, 0.f, 0.f, 0.f, 0.f, 0.f};
#pragma unroll
  for (int kb = 0; kb < DIM; kb += 4) {
    const int k = kb + 2 * hl;
    v2f a, b;
    a.x = Xs[m * XS_STRIDE + k];
    a.y = Xs[m * XS_STRIDE + k + 1];
    b.x = wp[k];
    b.y = wp[k + 1];
    // 8 args: (neg_a, A, neg_b, B, c_mod, C, reuse_a, reuse_b)
    acc = __builtin_amdgcn_wmma_f32_16x16x4_f32(false, a, false, b, (short)0,
                                                acc, false, false);
  }

  const float bv = bias[jcol];
#pragma unroll
  for (int r = 0; r < 8; ++r) {
    int node = nb + r + 8 * hl;
    if (node < N) {
      float val = acc[r] + bv;
      out[(size_t)node * DIM + jcol] = fmaxf(val, 0.0f);
    }
  }
}

// ---------------------------------------------------------------------------
extern "C" void kernel_launch(void* const* d_in, const int* in_sizes, int n_in,
                              void* d_out, int out_size, void* d_ws, size_t ws_size,
                              hipStream_t stream) {
  const float* v     = (const float*)d_in[0];
  const float* e     = (const float*)d_in[1];
  const int*   src   = (const int*)d_in[2];
  const int*   dst   = (const int*)d_in[3];
  const float* A_w   = (const float*)d_in[4];
  const float* A_b   = (const float*)d_in[5];
  const float* rs    = (const float*)d_in[6];
  const float* sigma = (const float*)d_in[7];
  const int N = in_sizes[0] / DIM;
  const int E = in_sizes[1];
  float* out = (float*)d_out;

  // workspace: f[E] | svf[N*DIM] | tmp[N*DIM]
  float* f   = (float*)d_ws;
  float* svf = f + (size_t)E;
  float* tmp = svf + (size_t)N * DIM;

  envelope_kernel<<<(E + 255) / 256, 256, 0, stream>>>(e, rs, sigma, f, E);

  const int nfeat = N * DIM;
  const int zgrid = (nfeat + 255) / 256;
  const int sgrid = (int)(((long long)E * 32 + 255) / 256);
  const int ggrid = (N + 15) / 16;

  const float* cur = v;
  float* nxt[3] = {out, tmp, out};   // v -> out -> tmp -> out (DEPTH=3)
  for (int d = 0; d < 3; ++d) {
    zero_kernel<<<zgrid, 256, 0, stream>>>(svf, nfeat);
    scatter_kernel<<<sgrid, 256, 0, stream>>>(cur, f, src, dst, svf, E);
    gemm_relu_kernel<<<ggrid, 256, 0, stream>>>(svf, v, A_w, A_b, nxt[d], N);
    cur = nxt[d];
  }
}